// NLI_Model_36713380446531
// MI455X (gfx1250) — compile-verified
//
#include <hip/hip_runtime.h>

// Masked segment pooling: max / mean / sum over time for [T,B,D] fp32.
// MI455X (gfx1250, wave32) bandwidth kernel:
//  - phase 1 streams the tensor via CDNA5 async DMA:
//      global_load_async_to_lds_b128 th:TH_LOAD_NT (double-buffered LDS
//      stages, counted s_wait_asynccnt, no barriers: waves consume only
//      what their own lanes loaded)
//  - s_wait_dscnt 0 before re-issuing into a buffer closes the
//    async-write vs in-flight ds_load reuse race
//  - skips all t >= lengths[b] (halves expected HBM traffic)
//  - non-temporal hints: 256 MB read once > 192 MB L2
//  - deterministic two-phase combine (no float atomics)

typedef float v4f __attribute__((ext_vector_type(4)));

static constexpr int T   = 2048;
static constexpr int B   = 64;
static constexpr int D   = 512;
static constexpr int D4  = D / 4;           // 128 float4 per (t,b) row
static constexpr int ROW4 = B * D4;         // float4 stride between t's
static constexpr unsigned ROW_BYTES = (unsigned)ROW4 * 16u; // 131072
static constexpr int ST  = 8;               // timesteps per async stage
static constexpr int MAXCHUNK = 32;

__device__ __forceinline__ v4f vmax4(v4f a, v4f b) {
    v4f r;
    r.x = fmaxf(a.x, b.x);
    r.y = fmaxf(a.y, b.y);
    r.z = fmaxf(a.z, b.z);
    r.w = fmaxf(a.w, b.w);
    return r;
}

// Issue one async stage: ST timesteps, 16 B per lane per step.
// goff0 = 32-bit global byte offset of this lane's column at the stage's t0;
// lds0  = LDS byte offset of this lane's slot in the stage buffer.
__device__ __forceinline__ void async_stage(const float* saddr, unsigned goff0,
                                            unsigned lds0) {
#pragma unroll
    for (int j = 0; j < ST; ++j) {
        asm volatile(
            "global_load_async_to_lds_b128 %0, %1, %2 th:TH_LOAD_NT"
            :
            : "v"(lds0 + (unsigned)j * 256u * 16u),
              "v"(goff0 + (unsigned)j * ROW_BYTES),
              "s"(saddr)
            : "memory");
    }
}

#define WAIT_ASYNC_ST()   asm volatile("s_wait_asynccnt 0x8" ::: "memory")
#define WAIT_ASYNC_ZERO() asm volatile("s_wait_asynccnt 0x0" ::: "memory")
#define WAIT_DS_ZERO()    asm volatile("s_wait_dscnt 0x0"    ::: "memory")

// ---------------- Phase 1: per-chunk partial sum/max ----------------
// grid = (B/2, nchunk), block = 256 (8 waves). Each wave covers one b
// (tid>>7 uniform per wave), each thread owns one float4 column.
__global__ __launch_bounds__(256) void pool_partial(
        const float* __restrict__ in, const int* __restrict__ lengths,
        float* __restrict__ ws_sum, float* __restrict__ ws_max,
        int chunk_len) {
    __shared__ v4f tile[2][ST][256];        // 64 KB double buffer

    const int tid   = threadIdx.x;
    const int b     = blockIdx.x * 2 + (tid >> 7);
    const int d4    = tid & (D4 - 1);
    const int chunk = blockIdx.y;

    const int L  = lengths[b];
    const int t0 = chunk * chunk_len;
    const int t1 = min(t0 + chunk_len, T);
    const int te = min(t1, L);              // wave-uniform; skip masked rows

    const unsigned col_bytes = ((unsigned)b * D4 + (unsigned)d4) * 16u;
    const unsigned ldsBase   = (unsigned)(uintptr_t)&tile[0][0][0]
                               + (unsigned)tid * 16u;

    v4f s = {0.0f, 0.0f, 0.0f, 0.0f};
    v4f m = {-__FLT_MAX__, -__FLT_MAX__, -__FLT_MAX__, -__FLT_MAX__};

    const int n     = te > t0 ? te - t0 : 0;
    const int nfull = n / ST;               // full async stages

    if (nfull > 0) {
        async_stage(in, (unsigned)t0 * ROW_BYTES + col_bytes, ldsBase);
        for (int st = 0; st < nfull; ++st) {
            const bool more = (st + 1) < nfull;
            if (more) {
                // ensure every prior ds_load of the buffer we are about to
                // overwrite has delivered its data (DScnt==0 is free here)
                WAIT_DS_ZERO();
                async_stage(in,
                            (unsigned)(t0 + (st + 1) * ST) * ROW_BYTES + col_bytes,
                            ldsBase + (unsigned)((st + 1) & 1) * (ST * 256u * 16u));
            }
            if (more) { WAIT_ASYNC_ST(); } else { WAIT_ASYNC_ZERO(); }
#pragma unroll
            for (int j = 0; j < ST; ++j) {
                v4f v = tile[st & 1][j][tid];
                s += v;
                m = vmax4(m, v);
            }
        }
    }

    // remainder (< ST timesteps): plain non-temporal b128 loads
    const v4f* __restrict__ base = (const v4f*)in + (size_t)b * D4 + d4;
    for (int t = t0 + nfull * ST; t < te; ++t) {
        v4f v = __builtin_nontemporal_load(base + (size_t)t * ROW4);
        s += v;
        m = vmax4(m, v);
    }

    const size_t o = ((size_t)chunk * B + b) * D4 + d4;
    ((v4f*)ws_sum)[o] = s;
    ((v4f*)ws_max)[o] = m;
}

// ---------------- Phase 2: combine partials, emit [3,B,D] ----------------
__global__ __launch_bounds__(256) void pool_final(
        const float* __restrict__ ws_sum, const float* __restrict__ ws_max,
        const int* __restrict__ lengths, float* __restrict__ out,
        int nchunk, int chunk_len) {
    const int tid = threadIdx.x;
    const int b   = blockIdx.x * 2 + (tid >> 7);
    const int d4  = tid & (D4 - 1);

    const int L    = lengths[b];
    const int used = min(nchunk, (L + chunk_len - 1) / chunk_len);

    v4f s = {0.0f, 0.0f, 0.0f, 0.0f};
    v4f m = {-__FLT_MAX__, -__FLT_MAX__, -__FLT_MAX__, -__FLT_MAX__};

    for (int c = 0; c < used; ++c) {
        const size_t o = ((size_t)c * B + b) * D4 + d4;
        s += ((const v4f*)ws_sum)[o];
        m  = vmax4(m, ((const v4f*)ws_max)[o]);
    }

    const float invL = 1.0f / (float)L;
    v4f mean = s * invL;

    v4f* o4 = (v4f*)out;
    const size_t col = (size_t)b * D4 + d4;
    o4[(size_t)0 * B * D4 + col] = m;     // pooled_max
    o4[(size_t)1 * B * D4 + col] = mean;  // pooled_mean
    o4[(size_t)2 * B * D4 + col] = s;     // pooled_sum
}

// ---------------- Fallback: single pass, no workspace ----------------
__global__ __launch_bounds__(256) void pool_direct(
        const float* __restrict__ in, const int* __restrict__ lengths,
        float* __restrict__ out) {
    const int tid = threadIdx.x;
    const int b   = blockIdx.x * 2 + (tid >> 7);
    const int d4  = tid & (D4 - 1);
    const int L   = lengths[b];

    const v4f* __restrict__ base = (const v4f*)in + (size_t)b * D4 + d4;

    v4f s = {0.0f, 0.0f, 0.0f, 0.0f};
    v4f m = {-__FLT_MAX__, -__FLT_MAX__, -__FLT_MAX__, -__FLT_MAX__};
#pragma unroll 4
    for (int t = 0; t < L; ++t) {
        v4f v = __builtin_nontemporal_load(base + (size_t)t * ROW4);
        s += v;
        m = vmax4(m, v);
    }
    const float invL = 1.0f / (float)L;
    v4f mean = s * invL;

    v4f* o4 = (v4f*)out;
    const size_t col = (size_t)b * D4 + d4;
    o4[(size_t)0 * B * D4 + col] = m;
    o4[(size_t)1 * B * D4 + col] = mean;
    o4[(size_t)2 * B * D4 + col] = s;
}

extern "C" void kernel_launch(void* const* d_in, const int* in_sizes, int n_in,
                              void* d_out, int out_size, void* d_ws, size_t ws_size,
                              hipStream_t stream) {
    const float* in      = (const float*)d_in[0];
    const int*   lengths = (const int*)d_in[1];
    float*       out     = (float*)d_out;

    const size_t per_chunk_bytes = (size_t)B * D * 2 * sizeof(float); // 256 KB
    size_t fit = ws_size / per_chunk_bytes;
    int nchunk = (fit > (size_t)MAXCHUNK) ? MAXCHUNK : (int)fit;

    if (nchunk >= 2) {
        const int chunk_len = (T + nchunk - 1) / nchunk;
        float* ws_sum = (float*)d_ws;
        float* ws_max = ws_sum + (size_t)nchunk * B * D;
        pool_partial<<<dim3(B / 2, nchunk), 256, 0, stream>>>(
            in, lengths, ws_sum, ws_max, chunk_len);
        pool_final<<<dim3(B / 2), 256, 0, stream>>>(
            ws_sum, ws_max, lengths, out, nchunk, chunk_len);
    } else {
        pool_direct<<<dim3(B / 2), 256, 0, stream>>>(in, lengths, out);
    }
}